// D4DispersionEnergy_26534307955291
// MI455X (gfx1250) — compile-verified
//
#include <hip/hip_runtime.h>
#include <hip/hip_bf16.h>
#include <math.h>

#define ZN 87
#define NREF 7
#define NC 5
#define NW 23
#define N_ATOMS 75000
#define N_PAIRS 1500000
#define NROW (ZN * NREF)   // 609
#define MPAD 640           // 40 tiles of 16
#define KPAD 24            // 23 -> 24 (one zero column)
#define NTILE (MPAD / 16)  // 40

#define INV_BOHR 1.8897261258369282f
#define HARTREE_F 27.211386024367243f
#define E3_F 20.085536923187668f
#define C6_SCALE 0.4774648292756860f   // 3/(2*pi)

typedef __attribute__((ext_vector_type(2))) float v2f;
typedef __attribute__((ext_vector_type(8))) float v8f;

__device__ __forceinline__ float softplus_f(float x) {
    return (x > 20.0f) ? x : log1pf(expf(x));
}

// ---------------- Phase 1: coordination numbers (pair scatter) ----------------
__global__ void k_pair_cn(const int* __restrict__ edge,
                          const float* __restrict__ lengths,
                          const int* __restrict__ species,
                          const float* __restrict__ rcov,
                          const float* __restrict__ en,
                          float* __restrict__ ncoord) {
    int p = blockIdx.x * blockDim.x + threadIdx.x;
    if (p >= N_PAIRS) return;
    int ii = edge[p];
    int jj = edge[N_PAIRS + p];
    float r = lengths[p] * INV_BOHR;
    int spi = species[ii], spj = species[jj];
    float rc = (4.0f / 3.0f) * (rcov[spi] + rcov[spj]);
    float d = fabsf(en[spi] - en[spj]) + 19.08857f;
    float den = 4.10451f * expf(-(d * d) * (1.0f / 254.5553148552f));
    float cf = den * 0.5f * (1.0f + erff(-7.5f * (r - rc) / rc));
    atomicAdd(&ncoord[ii], cf);
}

// ---------------- Phase 2: per-atom zeta --------------------------------------
__global__ void k_zeta(const int* __restrict__ species,
                       const float* __restrict__ ncoord,
                       const float* __restrict__ pq,
                       const float* __restrict__ scale_q_raw,
                       const float* __restrict__ zeff,
                       const float* __restrict__ gamv,
                       const float* __restrict__ refq,
                       const float* __restrict__ nmask,
                       const float* __restrict__ nweight,
                       const float* __restrict__ cnv,
                       float* __restrict__ zeta) {
    int i = blockIdx.x * blockDim.x + threadIdx.x;
    if (i >= N_ATOMS) return;
    int sp = species[i];
    float nc = ncoord[i];
    float gw[NREF];
    float norm = 0.0f;
#pragma unroll
    for (int a = 0; a < NREF; ++a) {
        int base = (sp * NREF + a) * NC;
        float g = 0.0f;
#pragma unroll
        for (int c = 0; c < NC; ++c) {
            float dd = nc - cnv[base + c];
            g += nmask[base + c] * expf(-6.0f * nweight[base + c] * dd * dd);
        }
        gw[a] = g;
        norm += g;
    }
    float inv = 1.0f / fmaxf(norm, 1e-7f);  // clip(.,1e-7) makes >1e-8 always true
    float spq = softplus_f(scale_q_raw[0]);
    float ze = zeff[sp];
    float gm = gamv[sp];
    float qmod = ze + pq[i];
#pragma unroll
    for (int a = 0; a < NREF; ++a) {
        float qref = ze + refq[sp * NREF + a] * spq;
        float zf;
        if (qmod > 1e-8f)
            zf = expf(3.0f * (1.0f - expf(2.0f * gm * (1.0f - qref / qmod))));
        else
            zf = E3_F;
        zeta[i * NREF + a] = zf * gw[a] * inv;
    }
}

// ---------------- Phase 3: B = alpha * sqrt(cpw), padded 640x24 ---------------
__global__ void k_bmat(const int* __restrict__ refsys,
                       const float* __restrict__ zeff,
                       const float* __restrict__ refh,
                       const float* __restrict__ sscale,
                       const float* __restrict__ secaiw,
                       const float* __restrict__ gamv,
                       const float* __restrict__ ascale,
                       const float* __restrict__ alphaiw,
                       const float* __restrict__ hcount,
                       const float* __restrict__ cpw,
                       const float* __restrict__ scale_q_raw,
                       float* __restrict__ Bm) {
    int row = blockIdx.x * blockDim.x + threadIdx.x;  // row = i*NREF + a
    if (row >= NROW) return;
    float spq = softplus_f(scale_q_raw[0]);
    int rs = refsys[row];
    float ze_r = zeff[rs];
    float ss_r = sscale[rs];
    float gm_r = gamv[rs];
    float qmod = ze_r + refh[row] * spq;
    float zfac;
    if (qmod > 1e-8f)
        zfac = expf(3.0f * (1.0f - expf(2.0f * gm_r * (1.0f - ze_r / qmod))));
    else
        zfac = E3_F;
    float asc = ascale[row];
    float hc = hcount[row];
#pragma unroll
    for (int w = 0; w < NW; ++w) {
        float sec = ss_r * secaiw[rs * NW + w] * zfac;
        float al = fmaxf(asc * (alphaiw[row * NW + w] - hc * sec), 0.0f);
        Bm[row * KPAD + w] = al * sqrtf(cpw[w]);
    }
}

// ---------------- Phase 4: refc6 = scale * B * B^T via V_WMMA_F32_16X16X4_F32 -
__global__ void k_refc6_wmma(const float* __restrict__ Bm,
                             float* __restrict__ C) {
    int wave = (blockIdx.x * blockDim.x + threadIdx.x) >> 5;
    int lane = threadIdx.x & 31;
    int tm = wave / NTILE;
    int tn = wave - tm * NTILE;
    int half = lane >> 4;
    int l16 = lane & 15;
    int rowA = tm * 16 + l16;
    int rowB = tn * 16 + l16;
#if defined(__gfx1250__) && __has_builtin(__builtin_amdgcn_wmma_f32_16x16x4_f32)
    v8f c = {0.f, 0.f, 0.f, 0.f, 0.f, 0.f, 0.f, 0.f};
#pragma unroll
    for (int kb = 0; kb < KPAD; kb += 4) {
        // ISA slot layout for 32-bit A 16x4: VGPR v, lane-half h -> k = v + 2h.
        // Same convention on both operands => K-reduction exact under any bijection.
        int k0 = kb + 2 * half;
        v2f a, b;
        a.x = Bm[rowA * KPAD + k0];
        a.y = Bm[rowA * KPAD + k0 + 1];
        b.x = Bm[rowB * KPAD + k0];
        b.y = Bm[rowB * KPAD + k0 + 1];
        c = __builtin_amdgcn_wmma_f32_16x16x4_f32(
            false, a, false, b, (short)0, c, false, false);
    }
#pragma unroll
    for (int r = 0; r < 8; ++r) {
        int m = tm * 16 + r + 8 * half;
        int n = tn * 16 + l16;
        C[m * MPAD + n] = c[r] * C6_SCALE;
    }
#else
    for (int r = 0; r < 8; ++r) {
        int m = tm * 16 + r + 8 * half;
        int n = tn * 16 + l16;
        float acc = 0.0f;
        for (int k = 0; k < KPAD; ++k)
            acc += Bm[m * KPAD + k] * Bm[n * KPAD + k];
        C[m * MPAD + n] = acc * C6_SCALE;
    }
#endif
}

// ---------------- Phase 5: pair energies + scatter ----------------------------
__global__ void k_pair_energy(const int* __restrict__ edge,
                              const float* __restrict__ lengths,
                              const int* __restrict__ species,
                              const float* __restrict__ zeta,
                              const float* __restrict__ refc6,
                              const float* __restrict__ sqrt_r4r2,
                              const float* __restrict__ s6_raw,
                              const float* __restrict__ s8_raw,
                              const float* __restrict__ a1_raw,
                              const float* __restrict__ a2_raw,
                              float* __restrict__ out) {
    int p = blockIdx.x * blockDim.x + threadIdx.x;
    if (p >= N_PAIRS) return;
    int ii = edge[p];
    int jj = edge[N_PAIRS + p];
    int spi = species[ii], spj = species[jj];

    float zi[NREF], zj[NREF];
#pragma unroll
    for (int a = 0; a < NREF; ++a) zi[a] = zeta[ii * NREF + a];
#pragma unroll
    for (int b = 0; b < NREF; ++b) zj[b] = zeta[jj * NREF + b];

    const float* blk = refc6 + (spi * NREF) * MPAD + spj * NREF;
    float c6 = 0.0f;
#pragma unroll
    for (int a = 0; a < NREF; ++a) {
        const float* rowp = blk + a * MPAD;
        float acc = 0.0f;
#pragma unroll
        for (int b = 0; b < NREF; ++b) acc += rowp[b] * zj[b];
        c6 += zi[a] * acc;
    }

    float r = lengths[p] * INV_BOHR;
    float rr = 3.0f * sqrt_r4r2[spi] * sqrt_r4r2[spj];
    float a1 = softplus_f(a1_raw[0]);
    float a2 = softplus_f(a2_raw[0]);
    float s6 = softplus_f(s6_raw[0]);
    float s8 = softplus_f(s8_raw[0]);
    float r0 = a1 * sqrtf(rr) + a2;
    float r0sq = r0 * r0;
    float r0_6 = r0sq * r0sq * r0sq;
    float r0_8 = r0_6 * r0sq;
    float r2 = r * r;
    float r6 = r2 * r2 * r2;
    float r8 = r6 * r2;
    float pair = -c6 * (s6 / (r6 + r0_6) + s8 * rr / (r8 + r0_8));
    atomicAdd(&out[ii], 0.5f * HARTREE_F * pair);
}

// ---------------- Launch ------------------------------------------------------
extern "C" void kernel_launch(void* const* d_in, const int* in_sizes, int n_in,
                              void* d_out, int out_size, void* d_ws, size_t ws_size,
                              hipStream_t stream) {
    const int*   species  = (const int*)d_in[0];
    const int*   edge     = (const int*)d_in[1];
    const float* lengths  = (const float*)d_in[2];
    const float* pq       = (const float*)d_in[3];
    const float* s6_raw   = (const float*)d_in[4];
    const float* s8_raw   = (const float*)d_in[5];
    const float* a1_raw   = (const float*)d_in[6];
    const float* a2_raw   = (const float*)d_in[7];
    const float* sq_raw   = (const float*)d_in[8];
    const int*   refsys   = (const int*)d_in[9];
    const float* zeff     = (const float*)d_in[10];
    const float* refh     = (const float*)d_in[11];
    const float* sscale   = (const float*)d_in[12];
    const float* secaiw   = (const float*)d_in[13];
    const float* gam      = (const float*)d_in[14];
    const float* ascale   = (const float*)d_in[15];
    const float* alphaiw  = (const float*)d_in[16];
    const float* hcount   = (const float*)d_in[17];
    const float* cpw      = (const float*)d_in[18];
    const float* rcov     = (const float*)d_in[19];
    const float* en       = (const float*)d_in[20];
    const float* nmask    = (const float*)d_in[21];
    const float* nweight  = (const float*)d_in[22];
    const float* cn       = (const float*)d_in[23];
    // d_in[24] fixgweights unused: norm is clipped to >=1e-7 so the >1e-8
    // branch in the reference is always taken.
    const float* refq     = (const float*)d_in[25];
    const float* r4r2     = (const float*)d_in[26];

    float* ws = (float*)d_ws;
    float* ncoord = ws;                           // 75000 floats
    float* zeta   = ws + 75008;                   // 525000 floats
    float* Bm     = ws + 75008 + 525056;          // 640*24 = 15360 floats
    float* refc6  = ws + 75008 + 525056 + 15360;  // 640*640 = 409600 floats

    hipMemsetAsync(ncoord, 0, (size_t)N_ATOMS * sizeof(float), stream);
    hipMemsetAsync(Bm, 0, (size_t)MPAD * KPAD * sizeof(float), stream);
    hipMemsetAsync(d_out, 0, (size_t)out_size * sizeof(float), stream);

    int pb = (N_PAIRS + 255) / 256;
    k_pair_cn<<<pb, 256, 0, stream>>>(edge, lengths, species, rcov, en, ncoord);

    int ab = (N_ATOMS + 255) / 256;
    k_zeta<<<ab, 256, 0, stream>>>(species, ncoord, pq, sq_raw, zeff, gam, refq,
                                   nmask, nweight, cn, zeta);

    k_bmat<<<(NROW + 63) / 64, 64, 0, stream>>>(refsys, zeff, refh, sscale, secaiw,
                                                gam, ascale, alphaiw, hcount, cpw,
                                                sq_raw, Bm);

    // 40x40 tiles, one wave32 per tile, 8 waves per 256-thread block -> 200 blocks
    k_refc6_wmma<<<(NTILE * NTILE) / 8, 256, 0, stream>>>(Bm, refc6);

    k_pair_energy<<<pb, 256, 0, stream>>>(edge, lengths, species, zeta, refc6, r4r2,
                                          s6_raw, s8_raw, a1_raw, a2_raw,
                                          (float*)d_out);
}